// NonLocalMeansSmoothing_64390149702239
// MI455X (gfx1250) — compile-verified
//
#include <hip/hip_runtime.h>

#define TILE  16
#define HALO  2
#define TDIM  (TILE + 2 * HALO)       // 20
#define NPIX  (TDIM * TDIM)           // 400
#define IMG   256
#define CCH   8

// ---- CDNA5 async global->LDS path (probe via __has_builtin, fallback keeps it compiling) ----
#if defined(__gfx1250__) && __has_builtin(__builtin_amdgcn_global_load_async_to_lds_b128)
#define NLM_ASYNC 1
typedef int v4i __attribute__((ext_vector_type(4)));
typedef __attribute__((address_space(1))) v4i GV4;    // global int4
typedef __attribute__((address_space(3))) v4i LV4;    // LDS int4
#else
#define NLM_ASYNC 0
#endif

__device__ __forceinline__ void copy16_to_lds(const float* g, float4* l) {
#if NLM_ASYNC
    // per-lane async DMA: copies 16 bytes from global to LDS, ASYNCcnt-tracked
    __builtin_amdgcn_global_load_async_to_lds_b128((GV4*)g, (LV4*)l, 0, 0);
#else
    *l = *(const float4*)g;
#endif
}

__device__ __forceinline__ void wait_async_copies() {
#if NLM_ASYNC
#if __has_builtin(__builtin_amdgcn_s_wait_asynccnt)
    __builtin_amdgcn_s_wait_asynccnt(0);
#else
    asm volatile("s_wait_asynccnt 0" ::: "memory");
#endif
#endif
}

__device__ __forceinline__ int refl(int p) {
    p = (p < 0) ? -p : p;                       // jnp 'reflect': -1 -> 1
    return (p >= IMG) ? (2 * IMG - 2 - p) : p;  // 256 -> 254
}

__global__ __launch_bounds__(256)
void nlm_smooth_kernel(const float* __restrict__ x, float* __restrict__ out) {
    // two channel planes: [20][20] float4 each; 16B pixel stride => conflict-free banking
    __shared__ float4 sp0[NPIX];
    __shared__ float4 sp1[NPIX];

    const int tj  = threadIdx.x;            // 0..15
    const int ti  = threadIdx.y;            // 0..15
    const int tid = ti * TILE + tj;
    const int b   = blockIdx.z;
    const int gi0 = blockIdx.y * TILE - HALO;
    const int gj0 = blockIdx.x * TILE - HALO;

    // ---- fill 20x20 halo tile (reflected gather) via async global->LDS ----
    for (int idx = tid; idx < NPIX; idx += 256) {
        const int r  = idx / TDIM;
        const int c  = idx - r * TDIM;
        const int gi = refl(gi0 + r);
        const int gj = refl(gj0 + c);
        const float* src = x + (((b * IMG + gi) * IMG) + gj) * CCH;
        copy16_to_lds(src,     &sp0[idx]);
        copy16_to_lds(src + 4, &sp1[idx]);
    }
    wait_async_copies();
    __syncthreads();

    // ---- per-pixel NLM ----
    const int cc = (ti + HALO) * TDIM + (tj + HALO);
    const float4 c0 = sp0[cc];
    const float4 c1 = sp1[cc];

    // unnormalized separable gaussian: exp(-(di^2+dj^2)/2); per-pixel w/sum(w) cancels scaling
    constexpr float ga[5] = {0.13533528f, 0.60653067f, 1.0f, 0.60653067f, 0.13533528f};

    float4 acc0 = make_float4(0.f, 0.f, 0.f, 0.f);
    float4 acc1 = make_float4(0.f, 0.f, 0.f, 0.f);
    float wsum = 0.f;

#pragma unroll
    for (int di = 0; di < 5; ++di) {
#pragma unroll
        for (int dj = 0; dj < 5; ++dj) {
            const int qi = (ti + di) * TDIM + (tj + dj);
            const float4 q0 = sp0[qi];
            const float4 q1 = sp1[qi];

            float t, d = 0.f;
            t = c0.x - q0.x; d = fmaf(t, t, d);
            t = c0.y - q0.y; d = fmaf(t, t, d);
            t = c0.z - q0.z; d = fmaf(t, t, d);
            t = c0.w - q0.w; d = fmaf(t, t, d);
            t = c1.x - q1.x; d = fmaf(t, t, d);
            t = c1.y - q1.y; d = fmaf(t, t, d);
            t = c1.z - q1.z; d = fmaf(t, t, d);
            t = c1.w - q1.w; d = fmaf(t, t, d);

            const float w = __expf(-d) * (ga[di] * ga[dj]);
            wsum += w;
            acc0.x = fmaf(w, q0.x, acc0.x);
            acc0.y = fmaf(w, q0.y, acc0.y);
            acc0.z = fmaf(w, q0.z, acc0.z);
            acc0.w = fmaf(w, q0.w, acc0.w);
            acc1.x = fmaf(w, q1.x, acc1.x);
            acc1.y = fmaf(w, q1.y, acc1.y);
            acc1.z = fmaf(w, q1.z, acc1.z);
            acc1.w = fmaf(w, q1.w, acc1.w);
        }
    }

    const float inv = 1.0f / wsum;   // wsum >= 1 (center term), never zero
    const int i = blockIdx.y * TILE + ti;
    const int j = blockIdx.x * TILE + tj;
    float* dst = out + (((b * IMG + i) * IMG) + j) * CCH;
    *(float4*)(dst)     = make_float4(acc0.x * inv, acc0.y * inv, acc0.z * inv, acc0.w * inv);
    *(float4*)(dst + 4) = make_float4(acc1.x * inv, acc1.y * inv, acc1.z * inv, acc1.w * inv);
}

extern "C" void kernel_launch(void* const* d_in, const int* in_sizes, int n_in,
                              void* d_out, int out_size, void* d_ws, size_t ws_size,
                              hipStream_t stream) {
    (void)n_in; (void)out_size; (void)d_ws; (void)ws_size;
    const float* x = (const float*)d_in[0];
    float* out = (float*)d_out;
    const int B = in_sizes[0] / (IMG * IMG * CCH);   // = 8
    dim3 grid(IMG / TILE, IMG / TILE, B);
    dim3 block(TILE, TILE, 1);
    nlm_smooth_kernel<<<grid, block, 0, stream>>>(x, out);
}